// TokenProto_41094247089000
// MI455X (gfx1250) — compile-verified
//
#include <hip/hip_runtime.h>

typedef float v2f __attribute__((ext_vector_type(2)));
typedef float v8f __attribute__((ext_vector_type(8)));

#define DDIM 1024
#define NLAB 64
#define STOK 16384   // 256*64 support tokens
#define QTOK 16384   // 256*64 query tokens

// workspace layout (bytes):
//   proto_acc : [64][1024] f32   @ 0        (262144)
//   cnt       : [64] int         @ 262144   (256)
//   p2        : [64] f32         @ 262400   (256)
//   protoB    : [256][4][32][2]f @ 262656   (262144)   total ~513 KB

__global__ __launch_bounds__(256) void k_zero(float* __restrict__ proto_acc,
                                              int* __restrict__ cnt) {
    int i = blockIdx.x * 256 + threadIdx.x;
    if (i < NLAB * DDIM) proto_acc[i] = 0.0f;
    if (i < NLAB) cnt[i] = 0;
}

// 128 blocks = 4 dim-slices x 32 token-slices, 256 threads.
// Each thread owns one dim column of the LDS accumulator -> no LDS contention.
__global__ __launch_bounds__(256) void k_accum(const float* __restrict__ s_emb,
                                               const int* __restrict__ tag,
                                               const int* __restrict__ msk,
                                               float* __restrict__ proto_acc,
                                               int* __restrict__ cnt) {
    __shared__ float lacc[NLAB * 256];   // 64 KB of the WGP's 320 KB LDS
    const int tid = threadIdx.x;
    const int ds  = blockIdx.x & 3;      // dim slice (4 x 256 dims)
    const int ts  = blockIdx.x >> 2;     // token slice (32 x 512 tokens)
    const int d   = ds * 256 + tid;

    #pragma unroll 8
    for (int i = 0; i < NLAB; ++i) lacc[i * 256 + tid] = 0.0f;
    __syncthreads();

    const int t0 = ts * (STOK / 32);
    const int t1 = t0 + (STOK / 32);
    for (int tok = t0; tok < t1; ++tok) {
        int   tg = tag[tok];                         // wave-uniform -> s_load
        int   mk = msk[tok];
        float v  = s_emb[(size_t)tok * DDIM + d];    // coalesced 128B/wave
        if (mk) lacc[tg * 256 + tid] += v;           // exclusive per thread
    }
    __syncthreads();

    #pragma unroll 8
    for (int i = 0; i < NLAB; ++i)
        atomicAdd(&proto_acc[i * DDIM + ds * 256 + tid], lacc[i * 256 + tid]);

    if (ds == 0) {  // counts: once per token slice
        for (int tok = t0 + tid; tok < t1; tok += 256)
            if (msk[tok]) atomicAdd(&cnt[tag[tok]], 1);
    }
}

// One block per label: normalize, reduce p2, and emit WMMA-B-fragment-ordered
// protoB so the GEMM's B loads are fully coalesced b64 loads.
// B frag layout (V_WMMA_F32_16X16X4_F32, 4x16 K-major): lane = (n&15)+16*h
// holds K = kc*4 + 2*h + {0,1} for column n.
__global__ __launch_bounds__(256) void k_finalize(const float* __restrict__ proto_acc,
                                                  const int* __restrict__ cnt,
                                                  float* __restrict__ p2,
                                                  float* __restrict__ protoB) {
    __shared__ float red[256];
    const int n   = blockIdx.x;
    const int tid = threadIdx.x;     // tid == kc (K chunk of 4)
    const float inv = 1.0f / fmaxf((float)cnt[n], 1.0f);

    float pv[4];
    float s = 0.0f;
    #pragma unroll
    for (int j = 0; j < 4; ++j) {
        float p = proto_acc[n * DDIM + tid * 4 + j] * inv;
        pv[j] = p;
        s += p * p;
    }
    red[tid] = s;
    __syncthreads();
    for (int off = 128; off > 0; off >>= 1) {
        if (tid < off) red[tid] += red[tid + off];
        __syncthreads();
    }
    if (tid == 0) p2[n] = red[0];

    const int nt = n >> 4;
    const int lm = n & 15;
    #pragma unroll
    for (int h = 0; h < 2; ++h) {
        int lane = lm + 16 * h;
        #pragma unroll
        for (int j = 0; j < 2; ++j)
            protoB[(size_t)(((tid * 4 + nt) * 32) + lane) * 2 + j] = pv[2 * h + j];
    }
}

// 128 blocks x 256 threads (8 waves). Each wave: 16 query rows x 64 protos.
// A frag (16x4 f32, ISA 7.12.2): lanes 0-15 row M=lane hold K0/K1,
// lanes 16-31 row M=lane-16 hold K2/K3 -> one v2f load per lane per chunk.
__global__ __launch_bounds__(256) void k_gemm(const float* __restrict__ q_emb,
                                              const int* __restrict__ q_mask,
                                              const float* __restrict__ protoB,
                                              const float* __restrict__ p2,
                                              float* __restrict__ out) {
    const int tid  = threadIdx.x;
    const int lane = tid & 31;
    const int w    = tid >> 5;
    const int lm   = lane & 15;
    const int lh   = lane >> 4;
    const int q0   = blockIdx.x * 128 + w * 16;

    const float* aptr = q_emb + (size_t)(q0 + lm) * DDIM + 2 * lh;
    const float* bptr = protoB + lane * 2;

    v8f acc0 = {}, acc1 = {}, acc2 = {}, acc3 = {};
    float sq = 0.0f;

    for (int kc = 0; kc < DDIM / 4; ++kc) {
        v2f a  = *(const v2f*)(aptr); aptr += 4;
        v2f b0 = *(const v2f*)(bptr + 0 * 64);
        v2f b1 = *(const v2f*)(bptr + 1 * 64);
        v2f b2 = *(const v2f*)(bptr + 2 * 64);
        v2f b3 = *(const v2f*)(bptr + 3 * 64);
        bptr += 256;
        acc0 = __builtin_amdgcn_wmma_f32_16x16x4_f32(false, a, false, b0, (short)0, acc0, false, false);
        acc1 = __builtin_amdgcn_wmma_f32_16x16x4_f32(false, a, false, b1, (short)0, acc1, false, false);
        acc2 = __builtin_amdgcn_wmma_f32_16x16x4_f32(false, a, false, b2, (short)0, acc2, false, false);
        acc3 = __builtin_amdgcn_wmma_f32_16x16x4_f32(false, a, false, b3, (short)0, acc3, false, false);
        sq += a.x * a.x + a.y * a.y;
    }

    // q2 for row lm: this lane held K = {0,1 mod 4}, partner lane^16 held {2,3 mod 4}
    sq += __shfl_xor(sq, 16);

    float p2v[4];
    #pragma unroll
    for (int nt = 0; nt < 4; ++nt) p2v[nt] = p2[nt * 16 + lm];

    v8f accs[4] = {acc0, acc1, acc2, acc3};
    // C/D layout: VGPR i -> (M=i, N=lm) for lanes 0-15, (M=i+8, N=lm) for 16-31
    #pragma unroll
    for (int i = 0; i < 8; ++i) {
        int   m    = i + 8 * lh;
        float q2m  = __shfl(sq, m);              // row m's |q|^2
        float mskv = (float)q_mask[q0 + m];
        #pragma unroll
        for (int nt = 0; nt < 4; ++nt) {
            float dot = accs[nt][i];
            out[(size_t)(q0 + m) * NLAB + nt * 16 + lm] =
                mskv * (2.0f * dot - q2m - p2v[nt]);
        }
    }
}

extern "C" void kernel_launch(void* const* d_in, const int* in_sizes, int n_in,
                              void* d_out, int out_size, void* d_ws, size_t ws_size,
                              hipStream_t stream) {
    const float* s_emb = (const float*)d_in[0];
    const int*   s_tag = (const int*)d_in[1];
    const int*   s_msk = (const int*)d_in[2];
    const float* q_emb = (const float*)d_in[3];
    const int*   q_msk = (const int*)d_in[4];
    float*       outp  = (float*)d_out;

    char*  ws        = (char*)d_ws;
    float* proto_acc = (float*)(ws);
    int*   cnt       = (int*)(ws + 262144);
    float* p2        = (float*)(ws + 262144 + 256);
    float* protoB    = (float*)(ws + 262144 + 512);

    hipLaunchKernelGGL(k_zero,     dim3((NLAB * DDIM + 255) / 256 + 1), dim3(256), 0, stream,
                       proto_acc, cnt);
    hipLaunchKernelGGL(k_accum,    dim3(128), dim3(256), 0, stream,
                       s_emb, s_tag, s_msk, proto_acc, cnt);
    hipLaunchKernelGGL(k_finalize, dim3(64),  dim3(256), 0, stream,
                       proto_acc, cnt, p2, protoB);
    hipLaunchKernelGGL(k_gemm,     dim3(128), dim3(256), 0, stream,
                       q_emb, q_msk, protoB, p2, outp);
}